// LightGCN_89069031784580
// MI455X (gfx1250) — compile-verified
//
#include <hip/hip_runtime.h>

// ---------------- problem constants (match reference) ----------------
#define NUM_USERS   100000
#define NUM_ITEMS   100000
#define EMB_DIM     64
#define NNZ_C       2000000
#define NN          (NUM_USERS + 1 + NUM_ITEMS)   // 200001 nodes
#define N_OUT_ROWS  (2 * (NUM_USERS + 1))         // 200002 output rows
#define TILES       (NNZ_C / 16)                  // 125000 (exact)
#define SCATTER_BLOCKS (TILES / 8)                // 15625 blocks of 8 waves

typedef __attribute__((ext_vector_type(2))) float v2f;
typedef __attribute__((ext_vector_type(8))) float v8f;

// node index -> row in the concatenated (user_out, item_out) output layout.
// nodes 0..100000 -> rows 0..100000 ; nodes 100001..200000 -> rows 100002..200001
__device__ __forceinline__ int out_row_of(int node) {
    return node + (node > NUM_USERS ? 1 : 0);
}

// ---------------------------------------------------------------------
// init: x0 = all_emb (concat), x1 = 0, d_out = all_emb in output layout
// float4-granular: q in [0, NN*16)
// ---------------------------------------------------------------------
__global__ void lgcn_init(const float* __restrict__ user,
                          const float* __restrict__ item,
                          float* __restrict__ x0, float* __restrict__ x1,
                          float* __restrict__ out) {
    int q = blockIdx.x * blockDim.x + threadIdx.x;
    if (q >= NN * 16) return;
    int node = q >> 4;
    int sub  = q & 15;
    const float4* src = (node <= NUM_USERS)
        ? (const float4*)user + (unsigned)node * 16u + sub
        : (const float4*)item + (unsigned)(node - NUM_USERS) * 16u + sub;
    float4 v = *src;
    ((float4*)x0)[q] = v;
    ((float4*)x1)[q] = make_float4(0.f, 0.f, 0.f, 0.f);
    ((float4*)out)[(unsigned)out_row_of(node) * 16u + sub] = v;
}

// ---------------------------------------------------------------------
// SpMM scatter via WMMA: one wave32 per 16-edge tile.
//   D(16x64) = diag(vals) * X[cols]   built from 4 chained
//   V_WMMA_F32_16X16X4_F32 per 16-dim chunk (exact f32 arithmetic),
//   then atomic-scattered into y[rows[e]] with global_atomic_add_f32.
// Edge metadata (rows/cols) is fetched with half-wave-uniform broadcast
// loads (one 64B cacheline per tile) instead of LDS permutes; all data
// offsets are unsigned 32-bit so the chunk displacement folds into the
// instruction immediate and loads/atomics use the SADDR form.
// Launched with EXACT grid (no guard) so EXEC is all ones at every WMMA.
// ---------------------------------------------------------------------
__global__ __launch_bounds__(256)
void lgcn_spmm_wmma(const int*   __restrict__ rows,
                    const int*   __restrict__ cols,
                    const float* __restrict__ vals,
                    const float* __restrict__ x,
                    float*       __restrict__ y) {
    const int lane = threadIdx.x & 31;
    const int wave = (blockIdx.x * blockDim.x + threadIdx.x) >> 5;
    const int base = wave * 16;
    const int m16  = lane & 15;   // N-coordinate / edge-within-tile
    const int hi   = lane >> 4;   // half-wave select

    // per-lane value for edge (base + m16); coalesced 16-wide
    const float v_lane = vals[base + m16];

    // Scatter rows: D-matrix VGPR g holds M = g + 8*hi -> rows[base+g+8*hi].
    // Consecutive per half-wave -> two b128 broadcast loads per lane.
    const int4* rp = (const int4*)(rows + base + 8 * hi);
    int4 ra = rp[0];
    int4 rb = rp[1];
    int rowv[8] = { ra.x, ra.y, ra.z, ra.w, rb.x, rb.y, rb.z, rb.w };

    // B-matrix gather columns per op j: VGPR0 carries K=2*hi, VGPR1 K=2*hi+1
    // -> pair cols[base + 4j + 2*hi], cols[base + 4j + 2*hi + 1]: int2 loads.
    unsigned cb0[4], cb1[4];   // precomputed element offsets col*64 + m16
#pragma unroll
    for (int j = 0; j < 4; ++j) {
        int2 cp = *(const int2*)(cols + base + 4 * j + 2 * hi);
        cb0[j] = (unsigned)cp.x * 64u + (unsigned)m16;
        cb1[j] = (unsigned)cp.y * 64u + (unsigned)m16;
    }

    // row base offsets (element units); nb displacement folds into IOFFSET
    unsigned rbse[8];
#pragma unroll
    for (int g = 0; g < 8; ++g)
        rbse[g] = (unsigned)rowv[g] * 64u + (unsigned)m16;

    // A matrices (16x4 f32, 2 VGPRs): lane holds M=m16, K=2*hi (.x), K=2*hi+1 (.y)
    // Op j implements rows m = 4j..4j+3 of diag(v): A_j[m][k] = v[4j+k]*(m==4j+k)
    v2f A[4];
#pragma unroll
    for (int j = 0; j < 4; ++j) {
        A[j].x = (m16 == 4 * j + 2 * hi)     ? v_lane : 0.0f;
        A[j].y = (m16 == 4 * j + 2 * hi + 1) ? v_lane : 0.0f;
    }

#pragma unroll
    for (int nb = 0; nb < EMB_DIM; nb += 16) {
        v8f c = {};   // chunk accumulator (16x16 f32)
#pragma unroll
        for (int j = 0; j < 4; ++j) {
            v2f b;
            b.x = x[cb0[j] + nb];
            b.y = x[cb1[j] + nb];
            // D[m][n] += sum_k A_j[m][k] * B_j[k][n]  (off-diag terms exactly 0)
            c = __builtin_amdgcn_wmma_f32_16x16x4_f32(
                    false, A[j], false, b, (short)0, c, false, false);
        }
        // D layout: VGPR g, lanes0-15 -> M=g, lanes16-31 -> M=g+8; N = m16
#pragma unroll
        for (int g = 0; g < 8; ++g)
            atomicAdd(&y[rbse[g] + nb], c[g]);
    }
}

// ---------------------------------------------------------------------
// accumulate layer result into d_out (output layout) and zero the buffer
// the next scatter will write into.
// ---------------------------------------------------------------------
__global__ void lgcn_acc_zero(const float* __restrict__ xs,
                              float* __restrict__ xz,
                              float* __restrict__ out) {
    int q = blockIdx.x * blockDim.x + threadIdx.x;
    if (q >= NN * 16) return;
    int node = q >> 4;
    int sub  = q & 15;
    float4 v = ((const float4*)xs)[q];
    float4* o = (float4*)out + (unsigned)out_row_of(node) * 16u + sub;
    float4 t = *o;
    t.x += v.x; t.y += v.y; t.z += v.z; t.w += v.w;
    *o = t;
    ((float4*)xz)[q] = make_float4(0.f, 0.f, 0.f, 0.f);
}

// ---------------------------------------------------------------------
// final: out = (out + last_layer) * 0.25, item row 0 (out row 100001) = 0
// ---------------------------------------------------------------------
__global__ void lgcn_final(const float* __restrict__ xs,
                           float* __restrict__ out) {
    int q = blockIdx.x * blockDim.x + threadIdx.x;
    if (q >= N_OUT_ROWS * 16) return;
    int orow = q >> 4;
    int sub  = q & 15;
    float4 res;
    if (orow == NUM_USERS + 1) {
        res = make_float4(0.f, 0.f, 0.f, 0.f);
    } else {
        int node = orow - (orow > NUM_USERS ? 1 : 0);
        float4 v = ((const float4*)xs)[(unsigned)node * 16u + sub];
        float4 t = ((const float4*)out)[q];
        res.x = (t.x + v.x) * 0.25f;
        res.y = (t.y + v.y) * 0.25f;
        res.z = (t.z + v.z) * 0.25f;
        res.w = (t.w + v.w) * 0.25f;
    }
    ((float4*)out)[q] = res;
}

// ---------------------------------------------------------------------
extern "C" void kernel_launch(void* const* d_in, const int* in_sizes, int n_in,
                              void* d_out, int out_size, void* d_ws, size_t ws_size,
                              hipStream_t stream) {
    const float* user_emb = (const float*)d_in[0];
    const float* item_emb = (const float*)d_in[1];
    const int*   adj_rows = (const int*)  d_in[2];
    const int*   adj_cols = (const int*)  d_in[3];
    const float* adj_vals = (const float*)d_in[4];
    float* out = (float*)d_out;

    float* x0 = (float*)d_ws;                       // 200001*64 f32 = 51.2 MB
    float* x1 = x0 + (size_t)NN * EMB_DIM;          // second ping-pong buffer

    const int EW_THREADS = 256;
    const int ew_blocks_nodes = (NN * 16 + EW_THREADS - 1) / EW_THREADS;
    const int ew_blocks_out   = (N_OUT_ROWS * 16 + EW_THREADS - 1) / EW_THREADS;

    // init: x0 = all_emb, x1 = 0, out = all_emb (acc)
    lgcn_init<<<ew_blocks_nodes, EW_THREADS, 0, stream>>>(user_emb, item_emb, x0, x1, out);

    // layer 1: x1 = A x0 ; out += x1 ; x0 = 0
    lgcn_spmm_wmma<<<SCATTER_BLOCKS, 256, 0, stream>>>(adj_rows, adj_cols, adj_vals, x0, x1);
    lgcn_acc_zero<<<ew_blocks_nodes, EW_THREADS, 0, stream>>>(x1, x0, out);

    // layer 2: x0 = A x1 ; out += x0 ; x1 = 0
    lgcn_spmm_wmma<<<SCATTER_BLOCKS, 256, 0, stream>>>(adj_rows, adj_cols, adj_vals, x1, x0);
    lgcn_acc_zero<<<ew_blocks_nodes, EW_THREADS, 0, stream>>>(x0, x1, out);

    // layer 3: x1 = A x0 ; out = (out + x1) / 4, item row 0 zeroed
    lgcn_spmm_wmma<<<SCATTER_BLOCKS, 256, 0, stream>>>(adj_rows, adj_cols, adj_vals, x0, x1);
    lgcn_final<<<ew_blocks_out, EW_THREADS, 0, stream>>>(x1, out);
}